// MinimumErrorRateLoss_60129542144015
// MI455X (gfx1250) — compile-verified
//
#include <hip/hip_runtime.h>
#include <hip/hip_bf16.h>
#include <math.h>

// Problem constants from the reference setup
#define B_  32
#define P_  16
#define H_  128
#define R_  100
#define C_  1024
#define BP_ (B_ * P_)       // 512
#define LMB_ 0.01f

typedef __attribute__((ext_vector_type(2))) float v2f;
typedef __attribute__((ext_vector_type(8))) float v8f;

#if __has_builtin(__builtin_amdgcn_update_dpp) && __has_builtin(__builtin_amdgcn_permlanex16)
#define FAST_XLANE 1
#else
#define FAST_XLANE 0
#endif

#if FAST_XLANE
// DPP move with row_xmask control (gfx10+ DPP16: 0x160 | xor_mask).
template <int CTRL>
__device__ __forceinline__ float dpp_mov_f(float x) {
  return __int_as_float(__builtin_amdgcn_update_dpp(
      __float_as_int(x), __float_as_int(x), CTRL, 0xf, 0xf, false));
}
// Swap the two 16-lane halves of a wave32 (identity lane selects).
__device__ __forceinline__ float permx16_f(float x) {
  return __int_as_float(__builtin_amdgcn_permlanex16(
      __float_as_int(x), __float_as_int(x), 0x76543210, 0xfedcba98, true, false));
}
__device__ __forceinline__ float wave_max32(float v) {
  v = fmaxf(v, dpp_mov_f<0x161>(v));   // xor 1
  v = fmaxf(v, dpp_mov_f<0x162>(v));   // xor 2
  v = fmaxf(v, dpp_mov_f<0x164>(v));   // xor 4
  v = fmaxf(v, dpp_mov_f<0x168>(v));   // xor 8
  v = fmaxf(v, permx16_f(v));          // xor 16
  return v;
}
__device__ __forceinline__ float xor16_add(float v) { return v + permx16_f(v); }
#else
__device__ __forceinline__ float wave_max32(float v) {
#pragma unroll
  for (int off = 16; off >= 1; off >>= 1)
    v = fmaxf(v, __shfl_xor(v, off, 32));
  return v;
}
__device__ __forceinline__ float xor16_add(float v) {
  return v + __shfl_xor(v, 16, 32);
}
#endif

// Kernel 1: per-(b,p) block. 8 waves, each wave owns 16 rows (h = wave*16+i).
// Per row: logsumexp over C=1024 via fp32 WMMA-with-ones row-sum, then gather
// logit@hyp (all h) and logit@ref (h<100). Writes log_probs[bp] and ce[bp].
__global__ __launch_bounds__(256) void k_lse_gather(
    const float* __restrict__ logits,
    const int*   __restrict__ ref,
    const int*   __restrict__ hyp,
    float* __restrict__ lp_out,
    float* __restrict__ ce_out) {
  const int bp   = blockIdx.x;           // 0..511
  const int lane = threadIdx.x & 31;
  const int wave = threadIdx.x >> 5;     // 0..7
  __shared__ float s_lp[8];
  __shared__ float s_ce[8];

  float lp_acc = 0.0f;
  float ce_acc = 0.0f;

#if __has_builtin(__builtin_amdgcn_wmma_f32_16x16x4_f32)
  const v2f ones = {1.0f, 1.0f};
#endif

  // Base row for this wave
  const size_t base = ((size_t)bp * H_ + wave * 16) * (size_t)C_;
  const float4* __restrict__ rp0 = (const float4*)(logits + base);

  // Prologue: load row 0 of this wave into registers (8 x global_load_b128).
  float4 cur[8];
#pragma unroll
  for (int it = 0; it < 8; ++it) cur[it] = rp0[lane + (it << 5)];

  for (int i = 0; i < 16; ++i) {
    const int h = wave * 16 + i;
    const size_t rowoff = base + (size_t)i * C_;

    // Software pipeline: issue next row's loads before reducing this row,
    // so ~16 b128 loads stay in flight per wave (s_wait_loadcnt 8 pattern).
    const int ni = (i < 15) ? (i + 1) : i;   // last iter: harmless cache re-read
    const float4* __restrict__ nrp = (const float4*)(logits + base + (size_t)ni * C_);
    float4 nxt[8];
#pragma unroll
    for (int it = 0; it < 8; ++it) nxt[it] = nrp[lane + (it << 5)];

    if (i < 14) {
      // prefetch row i+2 (one cacheline per lane covers the 4KB row)
      __builtin_prefetch((const void*)((const float4*)(logits + base + (size_t)(i + 2) * C_) + lane * 8), 0, 1);
    }

    // Pass 1: row max (pure-VALU DPP/permlane reduction, no LDS round-trips)
    float m = -INFINITY;
#pragma unroll
    for (int it = 0; it < 8; ++it)
      m = fmaxf(m, fmaxf(fmaxf(cur[it].x, cur[it].y), fmaxf(cur[it].z, cur[it].w)));
    m = wave_max32(m);

    // Pass 2: sum of exp(x - m)
    float total;
#if __has_builtin(__builtin_amdgcn_wmma_f32_16x16x4_f32)
    // Row-sum via matmul with ones: D = A x 1 + C accumulates every element
    // of A exactly once. 16 chained v_wmma_f32_16x16x4_f32 per row.
    v8f acc = {0.f, 0.f, 0.f, 0.f, 0.f, 0.f, 0.f, 0.f};
#pragma unroll
    for (int it = 0; it < 8; ++it) {
      const float ex = __expf(cur[it].x - m);
      const float ey = __expf(cur[it].y - m);
      const float ez = __expf(cur[it].z - m);
      const float ew = __expf(cur[it].w - m);
      v2f a0 = {ex, ey};
      v2f a1 = {ez, ew};
      acc = __builtin_amdgcn_wmma_f32_16x16x4_f32(false, a0, false, ones,
                                                  (short)0, acc, false, false);
      acc = __builtin_amdgcn_wmma_f32_16x16x4_f32(false, a1, false, ones,
                                                  (short)0, acc, false, false);
    }
    // D[0..7][n] lives in lanes 0..15 (n=lane), D[8..15][n] in lanes 16..31.
    float s = acc[0] + acc[1] + acc[2] + acc[3] + acc[4] + acc[5] + acc[6] + acc[7];
    total = xor16_add(s);
#else
    float s = 0.0f;
#pragma unroll
    for (int it = 0; it < 8; ++it) {
      s += __expf(cur[it].x - m) + __expf(cur[it].y - m) +
           __expf(cur[it].z - m) + __expf(cur[it].w - m);
    }
    s += __shfl_xor(s, 1, 32);  s += __shfl_xor(s, 2, 32);
    s += __shfl_xor(s, 4, 32);  s += __shfl_xor(s, 8, 32);
    total = xor16_add(s);
#endif
    const float lse = m + __logf(total);

    if (lane == 0) {
      const int htok = hyp[bp * H_ + h];
      lp_acc += logits[rowoff + htok] - lse;   // L0/L2 hit: row just streamed
      if (h < R_) {  // min_steps = min(R,H) = 100
        const int rtok = ref[bp * R_ + h];
        if (rtok != -100) {
          int sr = rtok < 0 ? 0 : (rtok > (C_ - 1) ? (C_ - 1) : rtok);
          ce_acc -= (logits[rowoff + sr] - lse);
        }
      }
    }

    // rotate pipeline registers
#pragma unroll
    for (int it = 0; it < 8; ++it) cur[it] = nxt[it];
  }

  if (lane == 0) { s_lp[wave] = lp_acc; s_ce[wave] = ce_acc; }
  __syncthreads();
  if (threadIdx.x == 0) {
    float lp = 0.f, ce = 0.f;
#pragma unroll
    for (int w = 0; w < 8; ++w) { lp += s_lp[w]; ce += s_ce[w]; }
    lp_out[bp] = lp;
    ce_out[bp] = ce;
  }
}

// Kernel 2: Levenshtein distance per (b,p); one thread per problem, DP row in LDS.
__global__ __launch_bounds__(64) void k_editdist(
    const int* __restrict__ ref,
    const int* __restrict__ hyp,
    float* __restrict__ dist_out) {
  __shared__ int srow[64][R_ + 1];   // 64*101*4 = 25856 B, conflict-free stride
  const int t  = threadIdx.x;
  const int bp = blockIdx.x * 64 + t;
  int* row = srow[t];
  for (int j = 0; j <= R_; ++j) row[j] = j;          // j * DEL_COST
  const int* __restrict__ r  = ref + bp * R_;
  const int* __restrict__ hh = hyp + bp * H_;
  for (int i = 0; i < H_; ++i) {
    const int htok = hh[i];
    int diag = row[0];
    int left = i + 1;                                 // new[0] = old[0] + INS
    row[0] = left;
    for (int j = 1; j <= R_; ++j) {
      const int up  = row[j];
      const int sub = diag + ((r[j - 1] == htok) ? 0 : 1);
      int val = up + 1;
      const int l1 = left + 1;
      val = (l1 < val) ? l1 : val;
      val = (sub < val) ? sub : val;
      row[j] = val;
      left = val;
      diag = up;
    }
  }
  dist_out[bp] = (float)row[R_];
}

// Kernel 3: final loss assembly -> scalar. Deterministic serial reductions.
__global__ __launch_bounds__(512) void k_finalize(
    const float* __restrict__ lp,
    const float* __restrict__ ce,
    const float* __restrict__ dist,
    float* __restrict__ out) {
  __shared__ float s_lp[BP_];
  __shared__ float s_er[BP_];
  __shared__ float s_loss[BP_];
  const int t = threadIdx.x;     // == bp
  const int b = t >> 4;
  const float l = lp[t];
  const float e = dist[t] * (1.0f / (float)R_);
  s_lp[t] = l;
  s_er[t] = e;
  __syncthreads();
  float m = -INFINITY;
  for (int k = 0; k < P_; ++k) m = fmaxf(m, s_lp[b * P_ + k]);
  float den = 0.0f, ermean = 0.0f;
  for (int k = 0; k < P_; ++k) {
    den    += __expf(s_lp[b * P_ + k] - m);
    ermean += s_er[b * P_ + k];
  }
  ermean *= (1.0f / (float)P_);
  const float w = __expf(l - m) / den;
  s_loss[t] = (e - ermean) * w + LMB_ * ce[t];
  __syncthreads();
  if (t == 0) {
    float sum = 0.0f;
    for (int k = 0; k < BP_; ++k) sum += s_loss[k];
    out[0] = sum * (1.0f / (float)BP_);
  }
}

extern "C" void kernel_launch(void* const* d_in, const int* in_sizes, int n_in,
                              void* d_out, int out_size, void* d_ws, size_t ws_size,
                              hipStream_t stream) {
  (void)in_sizes; (void)n_in; (void)out_size; (void)ws_size;
  const float* logits = (const float*)d_in[0];
  const int*   ref    = (const int*)d_in[1];
  const int*   hyp    = (const int*)d_in[2];
  float* out = (float*)d_out;

  float* lp_ws   = (float*)d_ws;          // 512 floats
  float* ce_ws   = lp_ws + BP_;           // 512 floats
  float* dist_ws = lp_ws + 2 * BP_;       // 512 floats

  k_lse_gather<<<BP_, 256, 0, stream>>>(logits, ref, hyp, lp_ws, ce_ws);
  k_editdist<<<BP_ / 64, 64, 0, stream>>>(ref, hyp, dist_ws);
  k_finalize<<<1, BP_, 0, stream>>>(lp_ws, ce_ws, dist_ws, out);
}